// TensorProductConv_78056735638116
// MI455X (gfx1250) — compile-verified
//
#include <hip/hip_runtime.h>
#include <stdint.h>

#define THREADS 256
#define WAVES 8
#define GROUP 4                      // W rows gathered per TDM descriptor
#define WROW 160                     // floats per W row
#define GROUP_ELEMS (GROUP * WROW)   // 640 f32 = 2560 B per buffer
#define SCAN_THREADS 1024

typedef unsigned int uint32x4 __attribute__((ext_vector_type(4)));
typedef int          int32x8  __attribute__((ext_vector_type(8)));
typedef int          int32x4  __attribute__((ext_vector_type(4)));

#define SQRT3_INV 0.57735026918962576451f
#define SQRT2_INV 0.70710678118654752440f

// ---- TDM gather: DMA up to 4 rows of W (indexed by edge id) into LDS ----
static __device__ inline void tdm_gather_rows(const float* Wbase, uint32_t ldsByteAddr,
                                              int nIdx, int E, const int idx[GROUP]) {
    uint64_t ga = (uint64_t)(uintptr_t)Wbase;
    uint32x4 g0;
    g0[0] = 1u | (1u << 30) | (1u << 31);         // count=1 | 32-bit indices | gather_mode
    g0[1] = ldsByteAddr;                          // lds_addr (bytes)
    g0[2] = (uint32_t)(ga & 0xffffffffu);         // global_addr[31:0] (tensor base)
    g0[3] = (uint32_t)((ga >> 32) & 0x01ffffffu)  // global_addr[56:32]
          | (2u << 30);                           // type=2 ("image")
    int32x8 g1;
    g1[0] = (int)(2u << 16);                                   // data_size = 4 bytes
    g1[1] = (int)(((uint32_t)WROW & 0xffffu) << 16);           // tensor_dim0 = 160
    g1[2] = (int)(((uint32_t)E & 0xffffu) << 16);              // tensor_dim1[15:0]
    g1[3] = (int)((((uint32_t)E >> 16) & 0xffffu)              // tensor_dim1[31:16]
                  | ((uint32_t)WROW << 16));                   // tile_dim0 = 160
    g1[4] = (int)((uint32_t)nIdx & 0xffffu);                   // tile_dim1 = #valid indices
    g1[5] = WROW;                                              // tensor_dim0_stride lo
    g1[6] = 0;
    g1[7] = 0;
    int32x4 g2; g2[0] = idx[0]; g2[1] = idx[1]; g2[2] = idx[2]; g2[3] = idx[3];
    int32x4 g3 = {0, 0, 0, 0};                                 // row_index_4..7 unused
    int32x8 gz8 = {0, 0, 0, 0, 0, 0, 0, 0};
    __builtin_amdgcn_tensor_load_to_lds(g0, g1, g2, g3, gz8, 0);
}

// ---------------- CSR construction (int atomics only) ----------------
__global__ void zero_i32_kernel(int* __restrict__ p, int n) {
    int i  = blockIdx.x * blockDim.x + threadIdx.x;
    int st = gridDim.x * blockDim.x;
    for (; i < n; i += st) p[i] = 0;
}

__global__ void count_rows_kernel(const long long* __restrict__ rows,
                                  int* __restrict__ counts, int E) {
    int e  = blockIdx.x * blockDim.x + threadIdx.x;
    int st = gridDim.x * blockDim.x;
    for (; e < E; e += st) atomicAdd(&counts[(int)rows[e]], 1);
}

__global__ void scan_counts_kernel(const int* __restrict__ counts,
                                   int* __restrict__ rowStart,
                                   int* __restrict__ cursor, int n) {
    __shared__ int s[SCAN_THREADS];
    const int tid = threadIdx.x;
    int carry = 0;
    for (int base = 0; base < n; base += SCAN_THREADS) {
        const int i = base + tid;
        const int v = (i < n) ? counts[i] : 0;
        __syncthreads();
        s[tid] = v;
        __syncthreads();
        for (int off = 1; off < SCAN_THREADS; off <<= 1) {
            const int t = (tid >= off) ? s[tid - off] : 0;
            __syncthreads();
            s[tid] += t;
            __syncthreads();
        }
        if (i < n) {
            const int excl = carry + s[tid] - v;
            rowStart[i] = excl;
            cursor[i]   = excl;
        }
        carry += s[SCAN_THREADS - 1];
    }
    if (tid == 0) rowStart[n] = carry;
}

__global__ void scatter_edges_kernel(const long long* __restrict__ rows,
                                     int* __restrict__ cursor,
                                     int* __restrict__ edgeIdx, int E) {
    int e  = blockIdx.x * blockDim.x + threadIdx.x;
    int st = gridDim.x * blockDim.x;
    for (; e < E; e += st) {
        const int r = (int)rows[e];
        const int p = atomicAdd(&cursor[r], 1);
        edgeIdx[p] = e;
    }
}

// ---------------- Main: one wave per node, zero fp atomics ----------------
__global__ __launch_bounds__(THREADS)
void tpconv_gather_kernel(const float* __restrict__ X, const float* __restrict__ Y,
                          const float* __restrict__ W,
                          const long long* __restrict__ cols,
                          const int* __restrict__ rowStart,
                          const int* __restrict__ edgeIdx,
                          float* __restrict__ Z, int N, int E) {
    extern __shared__ float smem[];      // WAVES * 2 * GROUP_ELEMS floats (40 KB)
    const int lane = threadIdx.x & 31;
    const int wave = threadIdx.x >> 5;
    const int n = blockIdx.x * WAVES + wave;
    if (n >= N) return;                  // wave-uniform; no barriers in this kernel

    float* wbuf0 = smem + wave * (2 * GROUP_ELEMS);   // wave-private LDS slice

    const int lo      = rowStart[n];
    const int cnt     = rowStart[n + 1] - lo;
    const int nGroups = (cnt + GROUP - 1) / GROUP;

    float accP1 = 0.f, accP4 = 0.f;
    float accP2_0 = 0.f, accP2_1 = 0.f, accP2_2 = 0.f;
    float accP3_0 = 0.f, accP3_1 = 0.f, accP3_2 = 0.f;
    float accP5_0 = 0.f, accP5_1 = 0.f, accP5_2 = 0.f;

    int idxCur[GROUP] = {0, 0, 0, 0};
    int colCur[GROUP] = {0, 0, 0, 0};
    int curCnt = 0;
    if (nGroups > 0) {
        curCnt = cnt < GROUP ? cnt : GROUP;
        for (int j = 0; j < GROUP; ++j) {
            if (j < curCnt) {
                idxCur[j] = edgeIdx[lo + j];
                colCur[j] = (int)cols[idxCur[j]];
                __builtin_prefetch(X + (size_t)colCur[j] * 128 + lane * 4, 0, 3);
            }
        }
        tdm_gather_rows(W, (uint32_t)(uintptr_t)wbuf0, curCnt, E, idxCur);
    }

    int buf = 0;
    for (int g = 0; g < nGroups; ++g) {
        int nxtCnt = 0;
        int idxNxt[GROUP] = {0, 0, 0, 0};
        int colNxt[GROUP] = {0, 0, 0, 0};
        if (g + 1 < nGroups) {
            const int rem = cnt - (g + 1) * GROUP;
            nxtCnt = rem < GROUP ? rem : GROUP;
            for (int j = 0; j < GROUP; ++j) {
                if (j < nxtCnt) {
                    idxNxt[j] = edgeIdx[lo + (g + 1) * GROUP + j];
                    colNxt[j] = (int)cols[idxNxt[j]];
                    __builtin_prefetch(X + (size_t)colNxt[j] * 128 + lane * 4, 0, 3);
                }
            }
            tdm_gather_rows(W, (uint32_t)(uintptr_t)(wbuf0 + (buf ^ 1) * GROUP_ELEMS),
                            nxtCnt, E, idxNxt);
            __builtin_amdgcn_s_wait_tensorcnt((short)1);   // current group's DMA done
        } else {
            __builtin_amdgcn_s_wait_tensorcnt((short)0);
        }

        const float* wb = wbuf0 + buf * GROUP_ELEMS;
        for (int j = 0; j < curCnt; ++j) {
            const int    e = idxCur[j];
            const float4 y = reinterpret_cast<const float4*>(Y)[e];  // y0, y1[0..2]

            const float* xp = X + (size_t)colCur[j] * 128;   // L2-resident gather
            const float  x0 = xp[lane];
            const int    xi = 32 + 3 * lane;
            const float  a0 = xp[xi], a1 = xp[xi + 1], a2 = xp[xi + 2];

            const float* wp = wb + j * WROW;                 // staged by TDM gather
            const float w0 = wp[lane];
            const float w1 = wp[32 + lane];
            const float w2 = wp[64 + lane];
            const float w3 = wp[96 + lane];
            const float w4 = wp[128 + lane];

            // p1: w0 * x0 * y0
            accP1 = fmaf(w0 * x0, y.x, accP1);
            // p2: (w1*x0) outer y1
            const float t1 = w1 * x0;
            accP2_0 = fmaf(t1, y.y, accP2_0);
            accP2_1 = fmaf(t1, y.z, accP2_1);
            accP2_2 = fmaf(t1, y.w, accP2_2);
            // p3: w2 * y0 * x1
            const float s = w2 * y.x;
            accP3_0 = fmaf(s, a0, accP3_0);
            accP3_1 = fmaf(s, a1, accP3_1);
            accP3_2 = fmaf(s, a2, accP3_2);
            // p4: w3 * (x1 . y1) / sqrt(3)
            const float dot = fmaf(a0, y.y, fmaf(a1, y.z, a2 * y.w));
            accP4 = fmaf(w3 * SQRT3_INV, dot, accP4);
            // p5: w4 * cross(x1, y1) / sqrt(2)
            const float c0 = fmaf(a1, y.w, -(a2 * y.z));
            const float c1 = fmaf(a2, y.y, -(a0 * y.w));
            const float c2 = fmaf(a0, y.z, -(a1 * y.y));
            const float k  = w4 * SQRT2_INV;
            accP5_0 = fmaf(k, c0, accP5_0);
            accP5_1 = fmaf(k, c1, accP5_1);
            accP5_2 = fmaf(k, c2, accP5_2);
        }
        for (int j = 0; j < GROUP; ++j) { idxCur[j] = idxNxt[j]; colCur[j] = colNxt[j]; }
        curCnt = nxtCnt;
        buf ^= 1;
    }

    // Single non-atomic write of this node's 352-float row.
    float* zp = Z + (size_t)n * 352;
    zp[lane]               = accP1;
    zp[32  + 3 * lane + 0] = accP2_0;
    zp[32  + 3 * lane + 1] = accP2_1;
    zp[32  + 3 * lane + 2] = accP2_2;
    zp[128 + 3 * lane + 0] = accP3_0;
    zp[128 + 3 * lane + 1] = accP3_1;
    zp[128 + 3 * lane + 2] = accP3_2;
    zp[224 + lane]         = accP4;
    zp[256 + 3 * lane + 0] = accP5_0;
    zp[256 + 3 * lane + 1] = accP5_1;
    zp[256 + 3 * lane + 2] = accP5_2;
}

extern "C" void kernel_launch(void* const* d_in, const int* in_sizes, int n_in,
                              void* d_out, int out_size, void* d_ws, size_t ws_size,
                              hipStream_t stream) {
    const float*     X    = (const float*)d_in[0];
    const float*     Y    = (const float*)d_in[1];
    const float*     W    = (const float*)d_in[2];
    const long long* rows = (const long long*)d_in[3];
    const long long* cols = (const long long*)d_in[4];
    float*           Z    = (float*)d_out;

    const int N = in_sizes[0] / 128;   // X is (N, 128)
    const int E = in_sizes[1] / 4;     // Y is (E, 4)

    // Workspace layout (ints): counts[N] | rowStart[N+1] | cursor[N] | edgeIdx[E]
    int* wsI      = (int*)d_ws;
    int* counts   = wsI;
    int* rowStart = wsI + N;
    int* cursor   = wsI + 2 * N + 1;
    int* edgeIdx  = wsI + 3 * N + 1;

    int zb = (N + THREADS - 1) / THREADS; if (zb > 2048) zb = 2048;
    zero_i32_kernel<<<zb, THREADS, 0, stream>>>(counts, N);

    int eb = (E + THREADS - 1) / THREADS; if (eb > 2048) eb = 2048;
    count_rows_kernel<<<eb, THREADS, 0, stream>>>(rows, counts, E);
    scan_counts_kernel<<<1, SCAN_THREADS, 0, stream>>>(counts, rowStart, cursor, N);
    scatter_edges_kernel<<<eb, THREADS, 0, stream>>>(rows, cursor, edgeIdx, E);

    const size_t shmem = (size_t)WAVES * 2 * GROUP_ELEMS * sizeof(float); // 40 KB
    const int blocks = (N + WAVES - 1) / WAVES;
    tpconv_gather_kernel<<<blocks, THREADS, shmem, stream>>>(
        X, Y, W, cols, rowStart, edgeIdx, Z, N, E);
}